// EncoderGRU_893353198039
// MI455X (gfx1250) — compile-verified
//
#include <hip/hip_runtime.h>
#include <hip/hip_bf16.h>

// CDNA5 / gfx1250 fp32 WMMA GRU encoder.
// D = A(16x4 f32) * B(4x16 f32) + C(16x16 f32), wave32.

#if defined(__HIP_DEVICE_COMPILE__)
#if !__has_builtin(__builtin_amdgcn_wmma_f32_16x16x4_f32)
#error "gfx1250 f32 WMMA builtin __builtin_amdgcn_wmma_f32_16x16x4_f32 not available"
#endif
#endif

typedef __attribute__((ext_vector_type(2))) float v2f;
typedef __attribute__((ext_vector_type(8))) float v8f;

#define TT 512
#define BB 64
#define EE 256
#define HH 256
#define TBROWS (TT * BB)

static __device__ __forceinline__ v8f wmma_f32(v2f a, v2f b, v8f c) {
#if defined(__HIP_DEVICE_COMPILE__)
  // (neg_a, A, neg_b, B, c_mod, C, reuse_a, reuse_b)
  return __builtin_amdgcn_wmma_f32_16x16x4_f32(false, a, false, b, (short)0, c,
                                               false, false);
#else
  return c;
#endif
}

static __device__ __forceinline__ float sigf(float x) {
  return 1.0f / (1.0f + __expf(-x));
}
static __device__ __forceinline__ float tanhfast(float x) {
  return 2.0f / (1.0f + __expf(-2.0f * x)) - 1.0f;
}

// ---------------------------------------------------------------------------
// Pack recurrent weights W[g][Dtot + k][n] (k,n in [0,256)) into WMMA B-frag
// order: Wpk[((g*64 + kt)*16 + nt)*32 + lane][2], where for lane L:
//   n = nt*16 + (L%16),  k = kt*4 + 2*(L/16) + v
// ---------------------------------------------------------------------------
__global__ __launch_bounds__(256) void pack_w_kernel(
    const float* __restrict__ W, float* __restrict__ Wpk, int Dtot, int wrows) {
  int idx = blockIdx.x * 256 + threadIdx.x;      // 0 .. 3*64*16*32-1
  if (idx >= 3 * 64 * 16 * 32) return;
  int lane = idx & 31;
  int r = idx >> 5;
  int nt = r & 15;  r >>= 4;
  int kt = r & 63;  r >>= 6;
  int g  = r;                                     // 0..2
  int n = nt * 16 + (lane & 15);
  int k = kt * 4 + 2 * (lane >> 4);
  size_t o = (size_t)idx * 2;
  size_t wbase = ((size_t)g * wrows + (Dtot + k)) * HH + n;
  Wpk[o + 0] = W[wbase];
  Wpk[o + 1] = W[wbase + HH];                     // k+1 row
}

// ---------------------------------------------------------------------------
// Input projection GEMM:  P[row][g*256+n] = sum_k A[row][k]*W[g][k][n] + b[g][n]
// A row = concat( A0[map0?map0[row]:row][0:K0], A1[map1?map1[row]:row][0:K1] )
// Block tile: 64(M) x 128(N), 8 waves, each wave 1 m-tile x 4 n-tiles.
// sB is pair-major: sBp[kpair][col][2] -> one ds_load_b64 per B fragment.
// Stride 288 dwords (== 32 mod 64) => 32 lanes x 2 dwords cover all 64 banks.
// ---------------------------------------------------------------------------
#define SBP_STRIDE 288

__global__ __launch_bounds__(256) void proj_gemm_kernel(
    float* __restrict__ P, const float* __restrict__ A0,
    const float* __restrict__ A1, const int* __restrict__ map0,
    const int* __restrict__ map1, int K0, int K1,
    const float* __restrict__ W, const float* __restrict__ bias, int wrows) {
  __shared__ float sA[64][20];              // 16-wide k-chunk, stride 20
  __shared__ float sBp[8 * SBP_STRIDE];     // 8 k-pairs x 128 cols x 2

  const int tid  = threadIdx.x;
  const int lane = tid & 31;
  const int wave = tid >> 5;      // 0..7
  const int mt   = wave & 3;      // m-tile 0..3
  const int half = wave >> 2;     // n half 0..1
  const int lm   = lane & 15;
  const int lh   = lane >> 4;

  const int mBlock = blockIdx.x * 64;
  const int nBlock = blockIdx.y * 128;
  const int g   = nBlock / HH;    // gate (128-col blocks never straddle gates)
  const int ncb = nBlock % HH;
  const int K = K0 + K1;

  v8f acc[4] = {};

  const int ar  = tid >> 2;            // 0..63 : A stage row
  const int akq = (tid & 3) << 2;      // 0,4,8,12
  const int bkr = tid >> 4;            // 0..15 : B stage k-row
  const int bcq = (tid & 15) << 3;     // 0..120

  // Hoisted row gathers (uniform over the whole k-loop).
  const int grow = mBlock + ar;
  const int ra0 = map0 ? map0[grow] : grow;
  const int ra1 = map1 ? map1[grow] : grow;
  const float* aSrc0 = A0 + (size_t)ra0 * K0 + akq;
  const float* aSrc1 = A1 + (size_t)ra1 * K1 + akq;
  // B stage target: row kr -> sBp[(kr>>1)*288 + (kr&1)], col c at +2c
  float* bDst = &sBp[(bkr >> 1) * SBP_STRIDE + (bkr & 1)];
  const float* wRow = W + ((size_t)g * wrows + bkr) * HH + ncb + bcq;

  for (int kb = 0; kb < K; kb += 16) {
    {  // stage A (64x16); chunk is uniformly within one segment (K0%16==0)
      const float* src = (kb < K0) ? (aSrc0 + kb) : (aSrc1 + (kb - K0));
      float4 av = *(const float4*)src;
      sA[ar][akq + 0] = av.x; sA[ar][akq + 1] = av.y;
      sA[ar][akq + 2] = av.z; sA[ar][akq + 3] = av.w;
    }
    {  // stage B (16x128) into pair-major layout
      const float* wsrc = wRow + (size_t)kb * HH;
      float4 b0 = *(const float4*)wsrc;
      float4 b1 = *(const float4*)(wsrc + 4);
      bDst[(bcq + 0) * 2] = b0.x; bDst[(bcq + 1) * 2] = b0.y;
      bDst[(bcq + 2) * 2] = b0.z; bDst[(bcq + 3) * 2] = b0.w;
      bDst[(bcq + 4) * 2] = b1.x; bDst[(bcq + 5) * 2] = b1.y;
      bDst[(bcq + 6) * 2] = b1.z; bDst[(bcq + 7) * 2] = b1.w;
    }
    __syncthreads();
    const int am = mt * 16 + lm;
#pragma unroll
    for (int nt = 0; nt < 4; ++nt) {
      const int bc = half * 64 + nt * 16 + lm;
#pragma unroll
      for (int kk4 = 0; kk4 < 16; kk4 += 4) {
        const int k0 = kk4 + (lh << 1);
        const int kp = (kk4 >> 1) + lh;          // k-pair index
        v2f a;
        a.x = sA[am][k0]; a.y = sA[am][k0 + 1];
        v2f b = *(const v2f*)(&sBp[kp * SBP_STRIDE + bc * 2]);
        acc[nt] = wmma_f32(a, b, acc[nt]);
      }
    }
    __syncthreads();
  }
  // epilogue: C layout lane L vgpr v -> m = v + 8*(L/16), n = L%16
#pragma unroll
  for (int nt = 0; nt < 4; ++nt) {
    int colg = ncb + half * 64 + nt * 16 + lm;
    float bval = bias[g * HH + colg];
#pragma unroll
    for (int v = 0; v < 8; ++v) {
      int row = mBlock + mt * 16 + v + (lh << 3);
      P[(size_t)row * 768 + g * HH + colg] = acc[nt][v] + bval;
    }
  }
}

// ---------------------------------------------------------------------------
// GRU recurrent scan. Block = 16 batch rows x full H=256; 16 waves, wave w
// owns n-tile [16w,16w+16) of all 3 gates. h kept in LDS (stride 260 so the
// 32-lane x 2-dword A-frag b64 reads hit all 64 banks once) + per-lane regs.
// Weights pre-packed in B-frag order, streamed from L2 each step.
// ---------------------------------------------------------------------------
struct ScanDir {
  const float* P;     // [TB][768] projections (bias included)
  const float* Wpk;   // packed recurrent weights [3][64][16][32][2]
  float* Sout;        // state output base
  float* hFin;        // final hidden [B][H] slot
  int outStride;      // 256 (scratch) or 512 (d_out)
  int outColBase;     // 0 or 256
  int rev;            // 1: out row = (T-1-t)*B + b  (bwd of final layer)
};

__global__ __launch_bounds__(512) void gru_scan_kernel(ScanDir dA, ScanDir dB) {
  const ScanDir d = (blockIdx.y == 0) ? dA : dB;
  __shared__ float hbuf[16][260];
  __shared__ float rhbuf[16][260];

  const int tid  = threadIdx.x;
  const int lane = tid & 31;
  const int wave = tid >> 5;         // 0..15 -> n-tile
  const int nb   = wave << 4;
  const int lm   = lane & 15;
  const int lh   = lane >> 4;
  const int brow0 = blockIdx.x << 4; // 16 batch rows per block

  for (int i = tid; i < 16 * 260; i += 512) {
    (&hbuf[0][0])[i] = 0.f;
    (&rhbuf[0][0])[i] = 0.f;
  }
  __syncthreads();

  v8f hreg = {};
  const float* wfrag = d.Wpk + (size_t)wave * 64 + lane * 2;
  // gate g, k-tile kt fragment at: wfrag + (g*64 + kt)*1024

  for (int t = 0; t < TT; ++t) {
    // ---------------- phase A: r and z gates ----------------
    v8f ar0, az0, ar1 = {}, az1 = {};
#pragma unroll
    for (int v = 0; v < 8; ++v) {
      size_t row = (size_t)(t * BB + brow0 + v + (lh << 3));
      ar0[v] = d.P[row * 768 + 0 * HH + nb + lm];   // r gate (W[0])
      az0[v] = d.P[row * 768 + 2 * HH + nb + lm];   // z gate (W[2])
    }
#pragma unroll 4
    for (int kt = 0; kt < 64; kt += 2) {
      int k0 = (kt << 2) + (lh << 1);
      v2f a0, a1;
      a0.x = hbuf[lm][k0];     a0.y = hbuf[lm][k0 + 1];
      a1.x = hbuf[lm][k0 + 4]; a1.y = hbuf[lm][k0 + 5];
      v2f br0 = *(const v2f*)(wfrag + (size_t)(0 * 64 + kt) * 1024);
      v2f br1 = *(const v2f*)(wfrag + (size_t)(0 * 64 + kt + 1) * 1024);
      v2f bz0 = *(const v2f*)(wfrag + (size_t)(2 * 64 + kt) * 1024);
      v2f bz1 = *(const v2f*)(wfrag + (size_t)(2 * 64 + kt + 1) * 1024);
      ar0 = wmma_f32(a0, br0, ar0);
      az0 = wmma_f32(a0, bz0, az0);
      ar1 = wmma_f32(a1, br1, ar1);
      az1 = wmma_f32(a1, bz1, az1);
    }
    v8f r, z;
#pragma unroll
    for (int v = 0; v < 8; ++v) {
      r[v] = sigf(ar0[v] + ar1[v]);
      z[v] = sigf(az0[v] + az1[v]);
    }
#pragma unroll
    for (int v = 0; v < 8; ++v)
      rhbuf[v + (lh << 3)][nb + lm] = r[v] * hreg[v];
    __syncthreads();

    // ---------------- phase B: candidate + state update ----------------
    v8f ac0, ac1 = {};
#pragma unroll
    for (int v = 0; v < 8; ++v) {
      size_t row = (size_t)(t * BB + brow0 + v + (lh << 3));
      ac0[v] = d.P[row * 768 + 1 * HH + nb + lm];   // cand gate (W[1])
    }
#pragma unroll 4
    for (int kt = 0; kt < 64; kt += 2) {
      int k0 = (kt << 2) + (lh << 1);
      v2f a0, a1;
      a0.x = rhbuf[lm][k0];     a0.y = rhbuf[lm][k0 + 1];
      a1.x = rhbuf[lm][k0 + 4]; a1.y = rhbuf[lm][k0 + 5];
      v2f bc0 = *(const v2f*)(wfrag + (size_t)(1 * 64 + kt) * 1024);
      v2f bc1 = *(const v2f*)(wfrag + (size_t)(1 * 64 + kt + 1) * 1024);
      ac0 = wmma_f32(a0, bc0, ac0);
      ac1 = wmma_f32(a1, bc1, ac1);
    }
    const int trow = d.rev ? (TT - 1 - t) : t;
#pragma unroll
    for (int v = 0; v < 8; ++v) {
      float cand = tanhfast(ac0[v] + ac1[v]);
      hreg[v] = z[v] * hreg[v] + (1.f - z[v]) * cand;
    }
#pragma unroll
    for (int v = 0; v < 8; ++v) {
      int m = v + (lh << 3);
      hbuf[m][nb + lm] = hreg[v];
      size_t orow = (size_t)trow * BB + brow0 + m;
      d.Sout[orow * d.outStride + d.outColBase + nb + lm] = hreg[v];
    }
    __syncthreads();
  }
#pragma unroll
  for (int v = 0; v < 8; ++v) {
    int m = v + (lh << 3);
    d.hFin[(size_t)(brow0 + m) * HH + nb + lm] = hreg[v];
  }
}

// ---------------------------------------------------------------------------
extern "C" void kernel_launch(void* const* d_in, const int* in_sizes, int n_in,
                              void* d_out, int out_size, void* d_ws,
                              size_t ws_size, hipStream_t stream) {
  const float* x     = (const float*)d_in[0];
  const int*   ixm   = (const int*)d_in[1];
  const float* fw_W0 = (const float*)d_in[2];
  const float* fw_b0 = (const float*)d_in[3];
  const float* fw_W1 = (const float*)d_in[4];
  const float* fw_b1 = (const float*)d_in[5];
  const float* bw_W0 = (const float*)d_in[6];
  const float* bw_b0 = (const float*)d_in[7];
  const float* bw_W1 = (const float*)d_in[8];
  const float* bw_b1 = (const float*)d_in[9];
  float* out = (float*)d_out;

  float* ws = (float*)d_ws;
  const size_t Psz  = (size_t)TBROWS * 768;
  const size_t Ssz  = (size_t)TBROWS * HH;
  const size_t Wpsz = (size_t)3 * 64 * 16 * 32 * 2;
  float* Pf    = ws;
  float* Pb    = Pf + Psz;
  float* S0f   = Pb + Psz;
  float* S0b   = S0f + Ssz;
  float* wpk0f = S0b + Ssz;
  float* wpk0b = wpk0f + Wpsz;
  float* wpk1f = wpk0b + Wpsz;
  float* wpk1b = wpk1f + Wpsz;
  if (ws_size < (2 * Psz + 2 * Ssz + 4 * Wpsz) * sizeof(float)) return;

  // Pack recurrent (h-part) weights into WMMA B-fragment order.
  pack_w_kernel<<<384, 256, 0, stream>>>(fw_W0, wpk0f, EE, EE + HH);
  pack_w_kernel<<<384, 256, 0, stream>>>(bw_W0, wpk0b, EE, EE + HH);
  pack_w_kernel<<<384, 256, 0, stream>>>(fw_W1, wpk1f, 2 * HH, 3 * HH);
  pack_w_kernel<<<384, 256, 0, stream>>>(bw_W1, wpk1b, 2 * HH, 3 * HH);

  dim3 pg(TBROWS / 64, 768 / 128);
  // Layer 0 input projections (fwd: x, bwd: x gathered by ix_mapping).
  proj_gemm_kernel<<<pg, 256, 0, stream>>>(Pf, x, x, nullptr, nullptr, EE, 0,
                                           fw_W0, fw_b0, EE + HH);
  proj_gemm_kernel<<<pg, 256, 0, stream>>>(Pb, x, x, ixm, nullptr, EE, 0,
                                           bw_W0, bw_b0, EE + HH);

  const size_t fhOff = (size_t)TBROWS * 2 * HH;
  const size_t bhOff = fhOff + 2 * (size_t)BB * HH;

  // Layer 0 scans (fwd + bwd concurrently via gridDim.y).
  ScanDir f0{Pf, wpk0f, S0f, out + fhOff, HH, 0, 0};
  ScanDir b0{Pb, wpk0b, S0b, out + bhOff, HH, 0, 0};
  gru_scan_kernel<<<dim3(BB / 16, 2), 512, 0, stream>>>(f0, b0);

  // Layer 1 projections: fwd in = [S0f, S0b[ix]], bwd in = [S0b, S0f[ix]].
  proj_gemm_kernel<<<pg, 256, 0, stream>>>(Pf, S0f, S0b, nullptr, ixm, HH, HH,
                                           fw_W1, fw_b1, 3 * HH);
  proj_gemm_kernel<<<pg, 256, 0, stream>>>(Pb, S0b, S0f, nullptr, ixm, HH, HH,
                                           bw_W1, bw_b1, 3 * HH);

  // Layer 1 scans write straight into d_out:
  //   fwd -> rows p, cols [0,256);  bwd -> rows ix(p), cols [256,512).
  ScanDir f1{Pf, wpk1f, out, out + fhOff + (size_t)BB * HH, 2 * HH, 0, 0};
  ScanDir b1{Pb, wpk1b, out, out + bhOff + (size_t)BB * HH, 2 * HH, HH, 1};
  gru_scan_kernel<<<dim3(BB / 16, 2), 512, 0, stream>>>(f1, b1);
}